// Mk1_91036126806096
// MI455X (gfx1250) — compile-verified
//
#include <hip/hip_runtime.h>
#include <cstdint>

// Problem constants (from reference): B=64, T=4096, F3=192 -> F=64 per chunk,
// 3 chunks folded into 192 sequences; UNITS=3 -> 12 gates; dense 9->4.
#define B_    64
#define T_    4096
#define F3_   192
#define F_    64
#define NSEQ  192
#define G_    12   // 4*UNITS

typedef __attribute__((ext_vector_type(2))) float v2f;
typedef __attribute__((ext_vector_type(8))) float v8f;

__device__ __forceinline__ float sigmoidf(float x) {
    return 1.0f / (1.0f + __expf(-x));
}

// ---------------------------------------------------------------------------
// Kernel 1: time-parallel input projection  Z[s,t,0..11] = X[s,t,:] @ W
// One wave computes a 16(time) x 16(gate, zero-padded from 12) tile via 16
// chained V_WMMA_F32_16X16X4_F32 (K=64 total). f32 WMMA => numerically
// faithful to the f32 reference.
//
// A layout (32-bit A 16x4): lanes 0-15 hold M=lane  (VGPR0=K0, VGPR1=K1);
//                           lanes 16-31 hold M=lane-16 (VGPR0=K2, VGPR1=K3).
// B mirror-symmetric (lane = N). C/D: VGPR r = row r (lanes 0-15) / r+8.
//
// W is staged once per block into LDS, column-major with stride 66
// (bank = (2n + k) mod 64 -> conflict-free across the 16 N-lanes) and
// zero-padded to 16 columns, so the inner loop is branch-free:
// each lane preloads its B half-column into 16 v2f regs (ds_load_b64),
// then the K-loop is pure global_load_b64(A) + v_wmma.
// ---------------------------------------------------------------------------
__global__ void proj_kernel(const float* __restrict__ X,   // [B,T,F3]
                            const float* __restrict__ W,   // [F,12]
                            float* __restrict__ Z)         // [NSEQ,T,12]
{
    __shared__ float Wlds[16 * 66];            // [n][k], stride 66, ~4.2 KB

    for (int i = threadIdx.x; i < 16 * F_; i += blockDim.x) {
        const int n = i >> 6;                  // 0..15
        const int k = i & 63;                  // 0..63
        Wlds[n * 66 + k] = (n < G_) ? W[(size_t)k * G_ + n] : 0.0f;
    }
    __syncthreads();

    const int wave = (blockIdx.x * blockDim.x + threadIdx.x) >> 5;
    const int lane = threadIdx.x & 31;
    const int ntiles = NSEQ * (T_ / 16);       // 49152 (grid sized exactly)
    if (wave >= ntiles) return;

    const int s  = wave >> 8;                  // sequence 0..191
    const int t0 = (wave & 255) << 4;          // time tile start
    const int c  = s >> 6;                     // chunk 0..2
    const int b  = s & 63;                     // batch  0..63

    const bool hi = lane >= 16;
    const int  l  = lane & 15;
    const int  koff = hi ? 2 : 0;

    // Preload this lane's B half-column (N = l) into registers: one
    // ds_load_b64 per K-quad, no divergence.
    v2f bfrag[16];
#pragma unroll
    for (int kk = 0; kk < 16; ++kk) {
        const int k = 4 * kk + koff;
        bfrag[kk] = *(const v2f*)&Wlds[l * 66 + k];
    }

    // This lane's A row: X[b, t0+l, c*64 + k]  (contiguous, 256B-aligned base)
    const float* arow = X + ((size_t)b * T_ + (size_t)(t0 + l)) * F3_ + c * F_;

    v8f acc = {};
#pragma unroll
    for (int kk = 0; kk < 16; ++kk) {
        const int k = 4 * kk + koff;
        const v2f a = *(const v2f*)&arow[k];   // global_load_b64
        acc = __builtin_amdgcn_wmma_f32_16x16x4_f32(
                  /*neg_a=*/false, a, /*neg_b=*/false, bfrag[kk],
                  /*c_mod=*/(short)0, acc, /*reuse_a=*/false, /*reuse_b=*/false);
    }

    if (l < G_) {
        float* zbase = Z + ((size_t)s * T_ + t0) * G_ + l;
#pragma unroll
        for (int r = 0; r < 8; ++r) {
            const int m = r + (hi ? 8 : 0);    // time row within tile
            zbase[(size_t)m * G_] = acc[r];
        }
    }
}

// ---------------------------------------------------------------------------
// Kernel 2: sequential recurrence. One lane per sequence (192 lanes, 6 waves
// spread over 6 blocks -> 6 WGPs). All state in registers; streams 48 B/step.
// ---------------------------------------------------------------------------
__global__ void lstm_kernel(const float* __restrict__ Z,    // [NSEQ,T,12]
                            const float* __restrict__ U,    // [3,12]
                            const float* __restrict__ bias, // [12]
                            float* __restrict__ H)          // [NSEQ,T,3]
{
    const int s = blockIdx.x * blockDim.x + threadIdx.x;
    if (s >= NSEQ) return;

    float u[3][G_], bv[G_];
#pragma unroll
    for (int r = 0; r < 3; ++r)
#pragma unroll
        for (int g = 0; g < G_; ++g) u[r][g] = U[r * G_ + g];
#pragma unroll
    for (int g = 0; g < G_; ++g) bv[g] = bias[g];

    float h0 = 0.f, h1 = 0.f, h2 = 0.f;
    float c0 = 0.f, c1 = 0.f, c2 = 0.f;

    const float* zrow = Z + (size_t)s * T_ * G_;
    float*       hrow = H + (size_t)s * T_ * 3;

    for (int t = 0; t < T_; ++t) {
        __builtin_prefetch(zrow + 16 * G_, 0, 0);   // global_prefetch_b8

        const float4 z0 = *(const float4*)(zrow + 0);
        const float4 z1 = *(const float4*)(zrow + 4);
        const float4 z2 = *(const float4*)(zrow + 8);
        const float z[G_] = { z0.x, z0.y, z0.z, z0.w,
                              z1.x, z1.y, z1.z, z1.w,
                              z2.x, z2.y, z2.z, z2.w };
        float gate[G_];
#pragma unroll
        for (int g = 0; g < G_; ++g)
            gate[g] = sigmoidf(z[g] + bv[g]
                               + h0 * u[0][g] + h1 * u[1][g] + h2 * u[2][g]);

        // gate order i(0..2), f(3..5), g(6..8), o(9..11); activation=sigmoid
        c0 = gate[3] * c0 + gate[0] * gate[6];
        c1 = gate[4] * c1 + gate[1] * gate[7];
        c2 = gate[5] * c2 + gate[2] * gate[8];
        h0 = gate[9]  * sigmoidf(c0);
        h1 = gate[10] * sigmoidf(c1);
        h2 = gate[11] * sigmoidf(c2);

        hrow[0] = h0; hrow[1] = h1; hrow[2] = h2;
        zrow += G_;
        hrow += 3;
    }
}

// ---------------------------------------------------------------------------
// Kernel 3: dense head  y[b,t,:] = sigmoid(concat(h_c0,h_c1,h_c2) @ Wd + bd)
// One thread per (b,t); coalesced float4 stores.
// ---------------------------------------------------------------------------
__global__ void dense_kernel(const float* __restrict__ H,   // [NSEQ,T,3]
                             const float* __restrict__ Wd,  // [9,4]
                             const float* __restrict__ bd,  // [4]
                             float* __restrict__ Y)         // [B,T,4]
{
    const size_t idx = (size_t)blockIdx.x * blockDim.x + threadIdx.x;
    if (idx >= (size_t)B_ * T_) return;
    const int b = (int)(idx >> 12);
    const int t = (int)(idx & 4095);

    float hv[9];
#pragma unroll
    for (int c = 0; c < 3; ++c) {
        const float* hp = H + ((size_t)(c * B_ + b) * T_ + t) * 3;
        hv[c * 3 + 0] = hp[0];
        hv[c * 3 + 1] = hp[1];
        hv[c * 3 + 2] = hp[2];
    }

    float4 y;
    float* yo = &y.x;
#pragma unroll
    for (int j = 0; j < 4; ++j) {
        float a = bd[j];
#pragma unroll
        for (int uu = 0; uu < 9; ++uu) a += hv[uu] * Wd[uu * 4 + j];
        yo[j] = sigmoidf(a);
    }
    *(float4*)(Y + idx * 4) = y;
}

// ---------------------------------------------------------------------------
extern "C" void kernel_launch(void* const* d_in, const int* in_sizes, int n_in,
                              void* d_out, int out_size, void* d_ws, size_t ws_size,
                              hipStream_t stream) {
    const float* X    = (const float*)d_in[0];   // [64,4096,192]
    const float* W    = (const float*)d_in[1];   // [64,12]
    const float* U    = (const float*)d_in[2];   // [3,12]
    const float* bias = (const float*)d_in[3];   // [12]
    const float* Wd   = (const float*)d_in[4];   // [9,4]
    const float* bd   = (const float*)d_in[5];   // [4]
    float* Y = (float*)d_out;                    // [64,4096,4]

    float* Z = (float*)d_ws;                             // 192*4096*12 f32
    float* H = Z + (size_t)NSEQ * T_ * G_;               // 192*4096*3  f32

    // 1) projection: 49152 tiles, 8 waves/block
    proj_kernel<<<6144, 256, 0, stream>>>(X, W, Z);
    // 2) recurrence: 192 lanes across 6 WGPs
    lstm_kernel<<<6, 32, 0, stream>>>(Z, U, bias, H);
    // 3) dense + sigmoid epilogue
    dense_kernel<<<((size_t)B_ * T_ + 255) / 256, 256, 0, stream>>>(H, Wd, bd, Y);
}